// DeviceCheckerMseLoss_63926293233938
// MI455X (gfx1250) — compile-verified
//
#include <hip/hip_runtime.h>

// CDNA5 / gfx1250, wave32.
// Masked-MSE reduction: HBM-bound (512MB streamed @ 23.3 TB/s => ~23us floor).
// Cross-lane reduction uses V_WMMA_F32_16X16X4_F32 (fp32-exact) with B = ones.

typedef float v2f  __attribute__((ext_vector_type(2)));
typedef float v8f  __attribute__((ext_vector_type(8)));
typedef float f32x4 __attribute__((ext_vector_type(4)));

#define WAVES_PER_BLOCK 8
#define BLOCK_THREADS   256
#define GRID_BLOCKS     2048
#define F_DIM           512
#define ROW_STRIDE_X    128   // T*D = 8*16 floats between x[b,0,0] entries

// Reduce a per-lane fp32 value to the full-wave sum using the f32 WMMA.
// A-matrix 16x4 f32 layout: lanes 0-15 -> {K=0 (vgpr0), K=1 (vgpr1)},
// lanes 16-31 -> {K=2, K=3}. With a.y = 0 and B = all-ones:
//   D[m][n] = acc[m] + acc[m+16]  (for every n).
// D layout: lane L<16 holds M=0..7 in 8 vgprs; lane L>=16 holds M=8..15.
// Sum of the 8 D vgprs in lane L plus lane L^16 = total wave sum.
__device__ __forceinline__ float wave_reduce_wmma(float v) {
    v2f a; a.x = v;    a.y = 0.0f;
    v2f b; b.x = 1.0f; b.y = 1.0f;
    v8f c = {};
    v8f d = __builtin_amdgcn_wmma_f32_16x16x4_f32(
        /*neg_a=*/false, a, /*neg_b=*/false, b,
        /*c_mod=*/(short)0, c, /*reuse_a=*/false, /*reuse_b=*/false);
    float s = ((d[0] + d[1]) + (d[2] + d[3])) + ((d[4] + d[5]) + (d[6] + d[7]));
    s += __shfl_xor(s, 16, 32);
    return s;  // identical in all 32 lanes
}

__global__ void __launch_bounds__(BLOCK_THREADS)
masked_mse_partial(const float* __restrict__ x,
                   const float* __restrict__ y,
                   const float* __restrict__ yp,
                   const int*   __restrict__ n_valid,
                   float*       __restrict__ ws,
                   int B) {
    const int lane = threadIdx.x & 31;
    const int wave = threadIdx.x >> 5;

    float acc = 0.0f;

    // One wave per row; grid-stride over rows. 131072/(2048*8) = 8 rows/wave.
    for (int b = blockIdx.x * WAVES_PER_BLOCK + wave; b < B;
         b += gridDim.x * WAVES_PER_BLOCK) {
        // Uniform-address load: one request, broadcast to the wave.
        const int dev = (int)x[(size_t)b * ROW_STRIDE_X];
        const int th  = n_valid[dev];

        const f32x4* yrow = (const f32x4*)(y  + (size_t)b * F_DIM);
        const f32x4* prow = (const f32x4*)(yp + (size_t)b * F_DIM);

        const int fbase = lane * 4;
#pragma unroll
        for (int i = 0; i < 4; ++i) {
            // Streaming data (> L2), single use: non-temporal b128 loads.
            f32x4 a = __builtin_nontemporal_load(&yrow[i * 32 + lane]);
            f32x4 p = __builtin_nontemporal_load(&prow[i * 32 + lane]);
            const int f0 = fbase + i * 128;
            float d0 = (f0 + 0 < th) ? (a.x - p.x) : 0.0f;
            float d1 = (f0 + 1 < th) ? (a.y - p.y) : 0.0f;
            float d2 = (f0 + 2 < th) ? (a.z - p.z) : 0.0f;
            float d3 = (f0 + 3 < th) ? (a.w - p.w) : 0.0f;
            acc = fmaf(d0, d0, acc);
            acc = fmaf(d1, d1, acc);
            acc = fmaf(d2, d2, acc);
            acc = fmaf(d3, d3, acc);
        }
    }

    // Wave sum via fp32 WMMA (EXEC is all-ones here; no divergence above
    // survives past the loop for a full wave).
    float wsum = wave_reduce_wmma(acc);

    __shared__ float lds[WAVES_PER_BLOCK];
    if (lane == 0) lds[wave] = wsum;
    __syncthreads();
    if (threadIdx.x == 0) {
        float t = 0.0f;
#pragma unroll
        for (int i = 0; i < WAVES_PER_BLOCK; ++i) t += lds[i];
        ws[blockIdx.x] = t;  // deterministic per-block partial
    }
}

__global__ void __launch_bounds__(BLOCK_THREADS)
reduce_final(const float* __restrict__ ws, int n,
             float* __restrict__ out, float scale) {
    const int lane = threadIdx.x & 31;
    const int wave = threadIdx.x >> 5;

    float acc = 0.0f;
    for (int i = threadIdx.x; i < n; i += BLOCK_THREADS) acc += ws[i];

    float wsum = wave_reduce_wmma(acc);

    __shared__ float lds[WAVES_PER_BLOCK];
    if (lane == 0) lds[wave] = wsum;
    __syncthreads();
    if (threadIdx.x == 0) {
        float t = 0.0f;
#pragma unroll
        for (int i = 0; i < WAVES_PER_BLOCK; ++i) t += lds[i];
        out[0] = t * scale;  // scale = 1/(B*F) = 2^-26 (exact)
    }
}

extern "C" void kernel_launch(void* const* d_in, const int* in_sizes, int n_in,
                              void* d_out, int out_size, void* d_ws, size_t ws_size,
                              hipStream_t stream) {
    const float* x  = (const float*)d_in[0];   // [B, T, D] f32 (device id at [:,0,0])
    const float* y  = (const float*)d_in[1];   // [B, F] f32
    const float* yp = (const float*)d_in[2];   // [B, F] f32
    const int*   nv = (const int*)d_in[3];     // [32] i32

    const int B = in_sizes[1] / F_DIM;         // 131072
    float* out = (float*)d_out;
    float* ws  = (float*)d_ws;                 // GRID_BLOCKS floats (8 KB)

    const float scale = 1.0f / ((float)B * (float)F_DIM);

    masked_mse_partial<<<GRID_BLOCKS, BLOCK_THREADS, 0, stream>>>(x, y, yp, nv, ws, B);
    reduce_final<<<1, BLOCK_THREADS, 0, stream>>>(ws, GRID_BLOCKS, out, scale);
}